// myWeightedBCEWithLogitsLoss_76184129897141
// MI455X (gfx1250) — compile-verified
//
#include <hip/hip_runtime.h>

typedef __attribute__((ext_vector_type(2))) float v2f;
typedef __attribute__((ext_vector_type(8))) float v8f;

#define BCE_BLOCKS 1024
#define BCE_BS     256
#define WAVES_PER_BLOCK (BCE_BS / 32)

// Full-wave (32-lane) f32 sum using V_WMMA_F32_16X16X4_F32.
// A: lane L puts its value in A-VGPR0, 0 in A-VGPR1.
//    Per ISA layout: A[m][0] = v[m] (lanes 0-15), A[m][2] = v[m+16] (lanes 16-31),
//    A[m][1] = A[m][3] = 0.
// B: all ones  =>  D[m][n] = v[m] + v[m+16]  (independent of n).
// D layout: lane L<16 holds D[0..7][L] in c[0..7]; lane L>=16 holds D[8..15][.].
// Sum of the 8 components gives half-sum; xor-16 shuffle completes it.
// EXEC must be all-ones at the call site (uniform control flow only).
__device__ __forceinline__ float wave_reduce_wmma(float v) {
    v2f a; a.x = v;    a.y = 0.0f;
    v2f b; b.x = 1.0f; b.y = 1.0f;
    v8f c = {};
    c = __builtin_amdgcn_wmma_f32_16x16x4_f32(
        /*neg_a=*/false, a, /*neg_b=*/false, b,
        /*c_mod=*/(short)0, c, /*reuse_a=*/false, /*reuse_b=*/false);
    float t = ((c[0] + c[1]) + (c[2] + c[3])) + ((c[4] + c[5]) + (c[6] + c[7]));
    t += __shfl_xor(t, 16, 32);
    return t; // total wave sum, broadcast to all 32 lanes
}

// Reference-faithful per-element loss:
//   x = sigmoid(img); loss = lbl*log(x+eps) + (1-lbl)*log(1-x+eps)
// Native-rate transcendentals (v_exp_f32 / v_log_f32) keep this HBM-bound.
__device__ __forceinline__ float bce_term(float img, float lbl) {
    const float eps = 1e-7f;
    float e  = __expf(-img);
    float r  = 1.0f / (1.0f + e);          // sigmoid; saturates cleanly at 0/1
    float lp = __logf(r + eps);
    float ln = __logf(1.0f - r + eps);     // 1-r form avoids inf*0 NaN
    return fmaf(lbl, lp - ln, ln);         // lbl*lp + (1-lbl)*ln
}

__global__ void bce_partial_kernel(const float* __restrict__ img,
                                   const float* __restrict__ lbl,
                                   float* __restrict__ part, int nvec) {
    __shared__ float wsum[WAVES_PER_BLOCK];
    const float4* img4 = (const float4*)img;
    const float4* lbl4 = (const float4*)lbl;
    const int tid    = threadIdx.x;
    const int stride = gridDim.x * BCE_BS;
    float acc = 0.0f;
    for (int i = blockIdx.x * BCE_BS + tid; i < nvec; i += stride) {
        float4 a = img4[i];   // global_load_b128
        float4 b = lbl4[i];   // global_load_b128
        acc += bce_term(a.x, b.x);
        acc += bce_term(a.y, b.y);
        acc += bce_term(a.z, b.z);
        acc += bce_term(a.w, b.w);
    }
    // Uniform control flow here: EXEC all-ones -> WMMA wave reduction.
    float ws = wave_reduce_wmma(acc);
    if ((tid & 31) == 0) wsum[tid >> 5] = ws;
    __syncthreads();
    if (tid == 0) {
        float s = 0.0f;
#pragma unroll
        for (int w = 0; w < WAVES_PER_BLOCK; ++w) s += wsum[w];
        part[blockIdx.x] = s;
    }
}

__global__ void bce_final_kernel(const float* __restrict__ part,
                                 float* __restrict__ out, int nb, float scale) {
    __shared__ float wsum[WAVES_PER_BLOCK];
    const int tid = threadIdx.x;
    float acc = 0.0f;
    for (int i = tid; i < nb; i += BCE_BS) acc += part[i];
    float ws = wave_reduce_wmma(acc);   // uniform control flow, EXEC all-ones
    if ((tid & 31) == 0) wsum[tid >> 5] = ws;
    __syncthreads();
    if (tid == 0) {
        float s = 0.0f;
#pragma unroll
        for (int w = 0; w < WAVES_PER_BLOCK; ++w) s += wsum[w];
        out[0] = -s * scale;
    }
}

extern "C" void kernel_launch(void* const* d_in, const int* in_sizes, int n_in,
                              void* d_out, int out_size, void* d_ws, size_t ws_size,
                              hipStream_t stream) {
    const float* img = (const float*)d_in[0];
    const float* lbl = (const float*)d_in[1];
    float* out  = (float*)d_out;
    float* part = (float*)d_ws;

    const int n    = in_sizes[0];      // 64*1*512*512 = 16,777,216
    const int nvec = n / 4;            // float4 elements

    int nb = BCE_BLOCKS;
    const int ws_cap = (int)(ws_size / sizeof(float));
    if (ws_cap > 0 && ws_cap < nb) nb = ws_cap;  // grid-stride handles any nb

    const float scale = 1.0f / (float)n;  // 1/(h*w*b*c) = 2^-24, exact

    bce_partial_kernel<<<nb, BCE_BS, 0, stream>>>(img, lbl, part, nvec);
    bce_final_kernel<<<1, BCE_BS, 0, stream>>>(part, out, nb, scale);
}